// ALIGNNConv_24051816858016
// MI455X (gfx1250) — compile-verified
//
#include <hip/hip_runtime.h>
#include <hip/hip_bf16.h>
#include <math.h>

// ---------------------------------------------------------------------------
// ALIGNN conv layer for MI455X (gfx1250), fp32 end-to-end.
// Dense GEMMs (K=64) run on V_WMMA_F32_16X16X4_F32 (fp32 matrix pipe):
// the workload is HBM-bound (~5-6 GB traffic vs ~60 GFLOP), so fp32 WMMA
// keeps reference precision with GEMM time far below memory time.
// N and the accumulate flag are template parameters so all GEMM operand
// addresses fold into immediate-offset global loads (no u64 address math
// in the wmma inner loop).
// ---------------------------------------------------------------------------

typedef __attribute__((ext_vector_type(2))) float v2f;
typedef __attribute__((ext_vector_type(8))) float v8f;

#define NN 50000L      // nodes
#define NE 600000L     // graph edges
#define NL 1200000L    // line-graph edges
#define BN_NBLK 512    // blocks for BN partial reduction
#define BN_EPS 1e-5f

// ---------------- workspace layout (floats) ----------------
static const long F_ASRC = 0;                       //  50000*128  node src proj
static const long F_ADST = F_ASRC + NN * 128;       //  50000*128  node dst proj
static const long F_M1   = F_ADST + NN * 128;       // 600000*128  edge msg pre-act  (later reused as B_src)
static const long F_MG   = F_M1   + NE * 128;       // 600000*64   gated msgs        (later reused as h2)
static const long F_H    = F_MG   + NE * 64;        //  50000*64   node aggregate
static const long F_MB   = F_H    + NN * 64;        // 600000*64   bottleneck pre-act
static const long F_Y2   = F_MB   + NE * 64;        // 600000*64   updated edge feats
static const long F_BDST = F_Y2   + NE * 64;        // 600000*128  lg dst proj       (later reused as m2 gated)
static const long F_M2   = F_BDST + NE * 128;       // 1200000*128 lg msg pre-act
static const long F_PART = F_M2   + NL * 128;       // BN partials
static const long F_SS   = F_PART + (long)BN_NBLK * 2 * 128; // [2C] scale,shift

// ---------------- device helpers ----------------
__device__ __forceinline__ float softplus_f(float x) {
  // numerically stable log1p(exp(x)) — matches jax.nn.softplus
  return fmaxf(x, 0.0f) + log1pf(__expf(-fabsf(x)));
}
__device__ __forceinline__ float sigmoid_f(float x) {
  return 1.0f / (1.0f + __expf(-x));
}

// ---------------- GEMM: out[M,N] (+)= A[M,64] @ W[64,N] (+ bias) ----------------
// One wave32 computes one 16x16 output tile via 16 chained fp32 WMMAs (K=64).
// A-operand layout (16x4 f32): lane r / r+16 hold row r, {K=2*half, 2*half+1}.
// B-operand layout (4x16 f32): lane r / r+16 hold col r, same K split.
// C/D layout (16x16 f32): vgpr i, lanes 0-15 -> (row i, col lane),
//                         lanes 16-31 -> (row i+8, col lane-16).
// N and ACC are compile-time so all loop loads use immediate offsets.
template <int N, bool ACC>
__global__ void alignn_gemm64_wmma(const float* __restrict__ A,
                                   const float* __restrict__ W,
                                   const float* __restrict__ bias,
                                   float* __restrict__ out,
                                   int M) {
  constexpr int tilesN = N >> 4;
  const int lane = threadIdx.x & 31;
  const int wv   = threadIdx.x >> 5;
  const int tile = blockIdx.x * 4 + wv;          // wave-uniform
  const int tilesTot = (M >> 4) * tilesN;
  if (tile >= tilesTot) return;                  // whole-wave exit: EXEC all-ones inside
  const int tn   = tile % tilesN;
  const int tm   = tile / tilesN;
  const int col0 = tn << 4;
  const int row0 = tm << 4;
  const int half = lane >> 4;
  const int r    = lane & 15;

  float* __restrict__ orow = out + (long)row0 * N + col0 + r + 8 * half * N;
  v8f c;
#pragma unroll
  for (int i = 0; i < 8; ++i) c[i] = ACC ? orow[(long)i * N] : 0.0f;

  const float* __restrict__ arow = A + (long)(row0 + r) * 64 + 2 * half;
  const float* __restrict__ wcol = W + col0 + r + 2 * half * N;
#pragma unroll
  for (int kk = 0; kk < 16; ++kk) {
    v2f a, b;
    a[0] = arow[kk * 4];             // immediate offsets (merged to b64)
    a[1] = arow[kk * 4 + 1];
    b[0] = wcol[kk * 4 * N];         // immediate offsets, constant k*N
    b[1] = wcol[(kk * 4 + 1) * N];
    c = __builtin_amdgcn_wmma_f32_16x16x4_f32(false, a, false, b,
                                              (short)0, c, false, false);
  }

  const float bv = bias ? bias[col0 + r] : 0.0f;
#pragma unroll
  for (int i = 0; i < 8; ++i) orow[(long)i * N] = c[i] + bv;
}

// ---------------- gather-add: E[e,:] += S[src[e],:] + D[dst[e],:]  (128 cols) ----
__global__ void alignn_gather_add(float* __restrict__ E,
                                  const float* __restrict__ S,
                                  const float* __restrict__ D,
                                  const int* __restrict__ src,
                                  const int* __restrict__ dst,
                                  long nE) {
  const long wave = ((long)blockIdx.x * blockDim.x + threadIdx.x) >> 5;
  const int lane = threadIdx.x & 31;
  if (wave >= nE) return;
  const long s = src[wave];
  const long d = dst[wave];
  const float4 a = ((const float4*)(S + s * 128))[lane];
  const float4 b = ((const float4*)(D + d * 128))[lane];
  float4* pe = (float4*)(E + wave * 128) + lane;
  float4 e = *pe;
  e.x += a.x + b.x; e.y += a.y + b.y; e.z += a.z + b.z; e.w += a.w + b.w;
  *pe = e;
}

// ---------------- BN pass 1: per-block column sums / sumsq ----------------
__global__ void alignn_bn_partial(const float* __restrict__ X, long M, int C,
                                  float* __restrict__ part) {
  const int tid  = threadIdx.x;           // blockDim = 256, C in {64,128}
  const int col  = tid & (C - 1);
  const int sub  = tid / C;
  const int nsub = 256 / C;
  float s = 0.0f, q = 0.0f;
  for (long rr = (long)blockIdx.x * nsub + sub; rr < M; rr += (long)gridDim.x * nsub) {
    const float v = X[rr * C + col];
    s += v; q += v * v;
  }
  __shared__ float shs[256];
  __shared__ float shq[256];
  shs[tid] = s; shq[tid] = q;
  __syncthreads();
  if (tid < C) {
    for (int k = 1; k < nsub; ++k) { s += shs[tid + k * C]; q += shq[tid + k * C]; }
    part[(long)blockIdx.x * 2 * C + tid]     = s;
    part[(long)blockIdx.x * 2 * C + C + tid] = q;
  }
}

// ---------------- BN pass 2: scale = g*rsqrt(var+eps), shift = b - mu*scale ----
__global__ void alignn_bn_finalize(const float* __restrict__ part, int nblocks, int C,
                                   const float* __restrict__ g,
                                   const float* __restrict__ b,
                                   float* __restrict__ ss, float invM) {
  const int col = threadIdx.x;            // blockDim = C
  float s = 0.0f, q = 0.0f;
  for (int i = 0; i < nblocks; ++i) {
    s += part[(long)i * 2 * C + col];
    q += part[(long)i * 2 * C + C + col];
  }
  const float mu  = s * invM;
  const float var = fmaxf(q * invM - mu * mu, 0.0f);
  const float sc  = g[col] * rsqrtf(var + BN_EPS);
  ss[col]     = sc;
  ss[C + col] = b[col] - mu * sc;
}

// ---------------- gate: mg[e,j] = sigmoid(bn(m[:,j])) * softplus(bn(m[:,64+j])) --
__global__ void alignn_gate(const float* __restrict__ mpre,  // [E,128]
                            const float* __restrict__ ss,    // [256] scale,shift
                            float* __restrict__ mg, long nE) {
  const long n = nE << 6;
  for (long i = (long)blockIdx.x * blockDim.x + threadIdx.x; i < n;
       i += (long)gridDim.x * blockDim.x) {
    const long e = i >> 6;
    const int  j = (int)(i & 63);
    const float f = fmaf(mpre[e * 128 + j],      ss[j],      ss[128 + j]);
    const float s = fmaf(mpre[e * 128 + 64 + j], ss[64 + j], ss[192 + j]);
    mg[i] = sigmoid_f(f) * softplus_f(s);
  }
}

// ---------------- zero fill ----------------
__global__ void alignn_zero(float* __restrict__ p, long n) {
  for (long i = (long)blockIdx.x * blockDim.x + threadIdx.x; i < n;
       i += (long)gridDim.x * blockDim.x)
    p[i] = 0.0f;
}

// ---------------- segment-sum scatter: h[dst[e],j] += mg[e,j] ----------------
__global__ void alignn_scatter(float* __restrict__ h,
                               const float* __restrict__ mg,
                               const int* __restrict__ dst, long nE) {
  const long n = nE << 6;
  for (long i = (long)blockIdx.x * blockDim.x + threadIdx.x; i < n;
       i += (long)gridDim.x * blockDim.x) {
    const long e = i >> 6;
    const int  j = (int)(i & 63);
    atomicAdd(&h[(long)dst[e] * 64 + j], mg[i]);
  }
}

// ---------------- out = softplus(base + v*scale + shift) ----------------
__global__ void alignn_out_softplus(const float* __restrict__ base,
                                    const float* __restrict__ v,
                                    const float* __restrict__ ss, // [128] (C=64)
                                    float* __restrict__ out, long n) {
  for (long i = (long)blockIdx.x * blockDim.x + threadIdx.x; i < n;
       i += (long)gridDim.x * blockDim.x) {
    const int j = (int)(i & 63);
    out[i] = softplus_f(base[i] + fmaf(v[i], ss[j], ss[64 + j]));
  }
}

// ---------------- out = base + softplus(v*scale + shift)  (bottleneck residual) --
__global__ void alignn_resid_softplus(const float* __restrict__ base,
                                      const float* __restrict__ v,
                                      const float* __restrict__ ss, // [128] (C=64)
                                      float* __restrict__ out, long n) {
  for (long i = (long)blockIdx.x * blockDim.x + threadIdx.x; i < n;
       i += (long)gridDim.x * blockDim.x) {
    const int j = (int)(i & 63);
    out[i] = base[i] + softplus_f(fmaf(v[i], ss[j], ss[64 + j]));
  }
}

// ---------------- host orchestration ----------------
static inline int grid1d(long n, int blk) {
  return (int)((n + blk - 1) / blk);
}

static void run_gemm128(const float* A, const float* W, const float* bias, float* out,
                        long M, hipStream_t stream) {
  const long tiles = (M >> 4) * 8;
  alignn_gemm64_wmma<128, false><<<(int)((tiles + 3) / 4), 128, 0, stream>>>(
      A, W, bias, out, (int)M);
}

static void run_gemm64(const float* A, const float* W, const float* bias, float* out,
                       long M, bool acc, hipStream_t stream) {
  const long tiles = (M >> 4) * 4;
  const int blocks = (int)((tiles + 3) / 4);
  if (acc)
    alignn_gemm64_wmma<64, true><<<blocks, 128, 0, stream>>>(A, W, bias, out, (int)M);
  else
    alignn_gemm64_wmma<64, false><<<blocks, 128, 0, stream>>>(A, W, bias, out, (int)M);
}

static void run_bn(const float* X, long M, int C, const float* g, const float* b,
                   float* part, float* ss, hipStream_t stream) {
  alignn_bn_partial<<<BN_NBLK, 256, 0, stream>>>(X, M, C, part);
  alignn_bn_finalize<<<1, C, 0, stream>>>(part, BN_NBLK, C, g, b, ss, 1.0f / (float)M);
}

extern "C" void kernel_launch(void* const* d_in, const int* in_sizes, int n_in,
                              void* d_out, int out_size, void* d_ws, size_t ws_size,
                              hipStream_t stream) {
  (void)in_sizes; (void)n_in; (void)out_size; (void)ws_size;

  // setup_inputs() dict order, params flattened leaf-by-leaf in insertion order:
  const float* x      = (const float*)d_in[0];   // [50000,64]
  const float* y      = (const float*)d_in[1];   // [600000,64]
  const float* z      = (const float*)d_in[2];   // [1200000,64]
  // params["node"]
  const float* nWsrc  = (const float*)d_in[3];   // [64,128]
  const float* nbsrc  = (const float*)d_in[4];
  const float* nWdst  = (const float*)d_in[5];
  const float* nbdst  = (const float*)d_in[6];
  const float* nWedge = (const float*)d_in[7];
  const float* nbedge = (const float*)d_in[8];
  const float* n_bnmg = (const float*)d_in[9];   // [128]
  const float* n_bnmb = (const float*)d_in[10];
  const float* n_bng  = (const float*)d_in[11];  // [64]
  const float* n_bnb  = (const float*)d_in[12];
  // params["edge"]
  const float* eWsrc  = (const float*)d_in[13];
  const float* ebsrc  = (const float*)d_in[14];
  const float* eWdst  = (const float*)d_in[15];
  const float* ebdst  = (const float*)d_in[16];
  const float* eWedge = (const float*)d_in[17];
  const float* ebedge = (const float*)d_in[18];
  const float* e_bnmg = (const float*)d_in[19];
  const float* e_bnmb = (const float*)d_in[20];
  const float* e_bng  = (const float*)d_in[21];
  const float* e_bnb  = (const float*)d_in[22];
  // bottleneck
  const float* Wb     = (const float*)d_in[23];  // [128,64]
  const float* bb     = (const float*)d_in[24];
  const float* bnbg   = (const float*)d_in[25];
  const float* bnbb   = (const float*)d_in[26];
  // indices
  const int* g_src  = (const int*)d_in[27];
  const int* g_dst  = (const int*)d_in[28];
  const int* lg_src = (const int*)d_in[29];
  const int* lg_dst = (const int*)d_in[30];

  float* ws   = (float*)d_ws;
  float* ASRC = ws + F_ASRC;
  float* ADST = ws + F_ADST;
  float* M1   = ws + F_M1;    // edge msg pre-act; reused as B_src in stage B
  float* MG   = ws + F_MG;    // gated msgs; reused as h2 in stage B
  float* H    = ws + F_H;
  float* MB   = ws + F_MB;
  float* Y2   = ws + F_Y2;
  float* BDST = ws + F_BDST;  // lg dst proj; reused as gated lg msgs
  float* M2   = ws + F_M2;
  float* PART = ws + F_PART;
  float* SS   = ws + F_SS;

  float* x_out = (float*)d_out;             // [50000,64]
  float* y_out = x_out + NN * 64;           // [600000,64]

  // ---------------- stage A: node CGCNN on g ----------------
  run_gemm128(x, nWsrc, nbsrc, ASRC, NN, stream);
  run_gemm128(x, nWdst, nbdst, ADST, NN, stream);
  run_gemm128(y, nWedge, nbedge, M1, NE, stream);
  alignn_gather_add<<<grid1d(NE * 32, 256), 256, 0, stream>>>(M1, ASRC, ADST,
                                                              g_src, g_dst, NE);
  run_bn(M1, NE, 128, n_bnmg, n_bnmb, PART, SS, stream);
  alignn_gate<<<grid1d(NE * 64, 256), 256, 0, stream>>>(M1, SS, MG, NE);
  alignn_zero<<<grid1d(NN * 64, 256), 256, 0, stream>>>(H, NN * 64);
  alignn_scatter<<<grid1d(NE * 64, 256), 256, 0, stream>>>(H, MG, g_dst, NE);
  run_bn(H, NN, 64, n_bng, n_bnb, PART, SS, stream);
  alignn_out_softplus<<<grid1d(NN * 64, 256), 256, 0, stream>>>(x, H, SS, x_out, NN * 64);

  // ---------------- edge bottleneck: cat(y, m) @ Wb = y@Wb[:64] + m@Wb[64:] ----
  run_gemm64(y,  Wb,           bb,      MB, NE, false, stream);
  run_gemm64(MG, Wb + 64 * 64, nullptr, MB, NE, true,  stream);
  run_bn(MB, NE, 64, bnbg, bnbb, PART, SS, stream);
  alignn_resid_softplus<<<grid1d(NE * 64, 256), 256, 0, stream>>>(y, MB, SS, Y2, NE * 64);

  // ---------------- stage B: edge CGCNN on line graph ----------------
  run_gemm128(Y2, eWsrc, ebsrc, M1,   NE, stream);   // B_src (reuse M1)
  run_gemm128(Y2, eWdst, ebdst, BDST, NE, stream);   // B_dst
  run_gemm128(z, eWedge, ebedge, M2,  NL, stream);
  alignn_gather_add<<<grid1d(NL * 32, 256), 256, 0, stream>>>(M2, M1, BDST,
                                                              lg_src, lg_dst, NL);
  run_bn(M2, NL, 128, e_bnmg, e_bnmb, PART, SS, stream);
  alignn_gate<<<grid1d(NL * 64, 256), 256, 0, stream>>>(M2, SS, BDST, NL); // mg2 -> BDST
  alignn_zero<<<grid1d(NE * 64, 256), 256, 0, stream>>>(MG, NE * 64);      // h2 -> MG
  alignn_scatter<<<grid1d(NL * 64, 256), 256, 0, stream>>>(MG, BDST, lg_dst, NL);
  run_bn(MG, NE, 64, e_bng, e_bnb, PART, SS, stream);
  alignn_out_softplus<<<grid1d(NE * 64, 256), 256, 0, stream>>>(Y2, MG, SS, y_out, NE * 64);
}